// GATCFConv_2774548873958
// MI455X (gfx1250) — compile-verified
//
#include <hip/hip_runtime.h>
#include <hip/hip_bf16.h>
#include <math.h>

// Problem dims (fixed by the reference)
#define BDIM 2
#define NDIM 1024
#define KDIM 64
#define GDIM 64
#define FDIM 128
#define HDIM 4
#define ODIM 128

#define SF_LD 68   // 64-wide tile, padded: bank = (4m + c) % 64, conflict-free
#define ST_LD 132  // 128-wide tile, padded likewise

typedef float v2f __attribute__((ext_vector_type(2)));
typedef float v8f __attribute__((ext_vector_type(8)));

__device__ __forceinline__ float sspf(float x) {
  // softplus(x) - log(2), numerically safe
  float sp = (x > 20.0f) ? x : log1pf(__expf(x));
  return sp - 0.69314718055994531f;
}

__device__ __forceinline__ v8f wmma4(v2f a, v2f b, v8f c) {
  // V_WMMA_F32_16X16X4_F32 : D = A(16x4) * B(4x16) + C(16x16), all fp32
  return __builtin_amdgcn_wmma_f32_16x16x4_f32(
      /*neg_a=*/false, a, /*neg_b=*/false, b,
      /*c_mod=*/(short)0, c, /*reuse_a=*/false, /*reuse_b=*/false);
}

// A fragment (16x4): lanes 0-15 -> K=k0,k0+1 ; lanes 16-31 -> K=k0+2,k0+3
__device__ __forceinline__ v2f load_afrag(const float* p, int row0, int ld,
                                          int k0, int lane) {
  int m  = lane & 15;
  int kk = (lane >> 4) << 1;
  const float* q = p + (size_t)(row0 + m) * ld + (k0 + kk);
  v2f a;
  a.x = q[0];
  a.y = q[1];
  return a;
}

// B fragment (4x16): row (K) striped across VGPRs, cols across lanes
__device__ __forceinline__ v2f load_bfrag(const float* p, int k0, int ld,
                                          int col0, int lane) {
  int nn = lane & 15;
  int kk = (lane >> 4) << 1;
  const float* q = p + (size_t)(k0 + kk) * ld + (col0 + nn);
  v2f b;
  b.x = q[0];
  b.y = q[ld];
  return b;
}

// C/D (16x16 f32): VGPR v -> row v (lanes 0-15) / row v+8 (lanes 16-31)
template <bool ACT, bool BIAS>
__device__ __forceinline__ void store_tile(float* dst, int ld, int row0,
                                           int col0, int lane, v8f c,
                                           float bias) {
  int mlo = (lane >> 4) << 3;
  int col = col0 + (lane & 15);
#pragma unroll
  for (int v = 0; v < 8; ++v) {
    float val = BIAS ? (c[v] + bias) : c[v];
    dst[(size_t)(row0 + v + mlo) * ld + col] = ACT ? sspf(val) : val;
  }
}

// Async global->LDS copy of 16 bytes per lane (ASYNCcnt-tracked DMA path)
__device__ __forceinline__ void async_g2l_b128(unsigned int lds_addr,
                                               const float* gaddr) {
  asm volatile("global_load_async_to_lds_b128 %0, %1, off"
               :
               : "v"(lds_addr), "v"(gaddr)
               : "memory");
}
__device__ __forceinline__ void wait_asynccnt0() {
  asm volatile("s_wait_asynccnt 0x0" ::: "memory");
}

// ---------------------------------------------------------------------------
// Kernel 1: y[b,h,n,f] = sum_i x[b,n,i] * w_in2f[h,i,f]
// grid 1024 x 128 threads; one 16x16 tile per wave (4096 tiles total)
// ---------------------------------------------------------------------------
__global__ void __launch_bounds__(128) k_proj(const float* __restrict__ x,
                                              const float* __restrict__ w_in2f,
                                              float* __restrict__ y) {
  int lane = threadIdx.x & 31;
  int wid  = blockIdx.x * 4 + (threadIdx.x >> 5);  // 0..4095
  int bh   = wid >> 9;                             // 0..7  (b*H + h)
  int rem  = wid & 511;
  int row0 = (rem >> 3) << 4;                      // 0..1008
  int col0 = (rem & 7) << 4;                       // 0..112
  int b    = bh >> 2;
  int h    = bh & 3;

  const float* A  = x + (size_t)b * NDIM * FDIM;
  const float* Bm = w_in2f + (size_t)h * FDIM * FDIM;

  v8f c = {};
#pragma unroll 4
  for (int kb = 0; kb < FDIM / 4; ++kb) {
    int k0 = kb * 4;
    v2f af = load_afrag(A, row0, FDIM, k0, lane);
    v2f bf = load_bfrag(Bm, k0, FDIM, col0, lane);
    c = wmma4(af, bf, c);
  }
  store_tile<false, false>(y + (size_t)bh * NDIM * FDIM, FDIM, row0, col0,
                           lane, c, 0.0f);
}

// ---------------------------------------------------------------------------
// Kernel 2: per-row attention scalars
//   a_c[row] = dot(y[row,:], w_att[h,:F]) ; a_n[row] = dot(y[row,:], w_att[h,F:])
// ---------------------------------------------------------------------------
__global__ void __launch_bounds__(128) k_att(const float* __restrict__ y,
                                             const float* __restrict__ w_att,
                                             float* __restrict__ a_c,
                                             float* __restrict__ a_n) {
  int lane = threadIdx.x & 31;
  int row  = blockIdx.x * 4 + (threadIdx.x >> 5);  // 0 .. B*H*N-1
  int h    = (row >> 10) & 3;

  const float* yr = y + (size_t)row * FDIM;
  const float* wa = w_att + (size_t)h * 2 * FDIM;

  float ac = 0.f, an = 0.f;
#pragma unroll
  for (int j = 0; j < 4; ++j) {
    int f = lane + j * 32;
    float v = yr[f];
    ac += v * wa[f];
    an += v * wa[FDIM + f];
  }
#pragma unroll
  for (int off = 16; off > 0; off >>= 1) {
    ac += __shfl_xor(ac, off, 32);
    an += __shfl_xor(an, off, 32);
  }
  if (lane == 0) {
    a_c[row] = ac;
    a_n[row] = an;
  }
}

// ---------------------------------------------------------------------------
// Kernel 3 (fused, per (b,n)): async-stage f_ij tile into LDS, softmax
// attention (overlapped with the DMA), 2-layer filter MLP via WMMA, then the
// gather-weighted aggregation over neighbors.
// ---------------------------------------------------------------------------
__global__ void __launch_bounds__(256) k_fused(
    const float* __restrict__ fij, const int* __restrict__ nbrs,
    const float* __restrict__ mask, const float* __restrict__ wf1,
    const float* __restrict__ bf1, const float* __restrict__ wf2,
    const float* __restrict__ bf2, const float* __restrict__ y,
    const float* __restrict__ a_c, const float* __restrict__ a_n,
    float* __restrict__ agg) {
  __shared__ __align__(16) float sF[KDIM * SF_LD];  // staged f_ij tile
  __shared__ __align__(16) float sT[KDIM * ST_LD];  // T1, then W
  __shared__ float sAtt[HDIM * KDIM];               // logits -> alpha
  __shared__ float sRed[2 * HDIM];                  // per-head max, 1/sum
  __shared__ int   sNb[KDIM];
  __shared__ float sMk[KDIM];
  __shared__ float sAgg[2 * FDIM];

  const int bn   = blockIdx.x;   // b*N + n
  const int b    = bn >> 10;
  const int n    = bn & 1023;
  const int tid  = threadIdx.x;
  const int lane = tid & 31;
  const int wave = tid >> 5;
  const int col0 = wave << 4;    // 0..112

  // ---- kick off async DMA of the 64x64 f_ij tile into padded LDS ----
  {
    const float* gbase = fij + (size_t)bn * KDIM * GDIM;
    unsigned int lds_base = (unsigned int)(uintptr_t)(void*)sF;
#pragma unroll
    for (int i = 0; i < 4; ++i) {
      int chunk = tid + i * 256;  // 0..1023 float4 chunks
      int r  = chunk >> 4;        // row 0..63
      int c4 = chunk & 15;        // float4 within row
      unsigned int laddr = lds_base + (unsigned int)((r * SF_LD + c4 * 4) * 4);
      async_g2l_b128(laddr, gbase + (r * GDIM + c4 * 4));
    }
  }

  // ---- attention logits (overlaps the async DMA) ----
  {
    int h  = tid >> 6;  // 0..3
    int k  = tid & 63;
    int nb = nbrs[(size_t)bn * KDIM + k];
    float e = sspf(a_c[(size_t)(b * HDIM + h) * NDIM + n] +
                   a_n[(size_t)(b * HDIM + h) * NDIM + nb]);
    sAtt[h * KDIM + k] = e;
    if (h == 0) {
      sNb[k] = nb;
      sMk[k] = mask[(size_t)bn * KDIM + k];
    }
  }
  __syncthreads();
  if (tid < HDIM) {
    float mx = -1e30f;
    for (int k = 0; k < KDIM; ++k) mx = fmaxf(mx, sAtt[tid * KDIM + k]);
    float sm = 0.f;
    for (int k = 0; k < KDIM; ++k) sm += __expf(sAtt[tid * KDIM + k] - mx);
    sRed[tid * 2 + 0] = mx;
    sRed[tid * 2 + 1] = 1.0f / sm;
  }
  __syncthreads();
  {
    int h = tid >> 6, k = tid & 63;
    sAtt[h * KDIM + k] =
        __expf(sAtt[h * KDIM + k] - sRed[h * 2]) * sRed[h * 2 + 1];
  }

  // ---- drain DMA, make tile visible to the whole block ----
  wait_asynccnt0();
  __syncthreads();

  // ---- GEMM1: T1 = ssp(f_ij(64x64) * wf1(64x128) + bf1) -> sT ----
  {
    v8f c0 = {}, c1 = {}, c2 = {}, c3 = {};
#pragma unroll 4
    for (int kb = 0; kb < GDIM / 4; ++kb) {
      int k0 = kb * 4;
      v2f bf = load_bfrag(wf1, k0, FDIM, col0, lane);
      v2f a0 = load_afrag(sF, 0, SF_LD, k0, lane);
      v2f a1 = load_afrag(sF, 16, SF_LD, k0, lane);
      v2f a2 = load_afrag(sF, 32, SF_LD, k0, lane);
      v2f a3 = load_afrag(sF, 48, SF_LD, k0, lane);
      c0 = wmma4(a0, bf, c0);
      c1 = wmma4(a1, bf, c1);
      c2 = wmma4(a2, bf, c2);
      c3 = wmma4(a3, bf, c3);
    }
    float bb = bf1[col0 + (lane & 15)];
    store_tile<true, true>(sT, ST_LD, 0, col0, lane, c0, bb);
    store_tile<true, true>(sT, ST_LD, 16, col0, lane, c1, bb);
    store_tile<true, true>(sT, ST_LD, 32, col0, lane, c2, bb);
    store_tile<true, true>(sT, ST_LD, 48, col0, lane, c3, bb);
  }
  __syncthreads();

  // ---- GEMM2: W = T1(64x128) * wf2(128x128) + bf2, overwrite sT ----
  {
    v8f c0 = {}, c1 = {}, c2 = {}, c3 = {};
#pragma unroll 4
    for (int kb = 0; kb < FDIM / 4; ++kb) {
      int k0 = kb * 4;
      v2f bf = load_bfrag(wf2, k0, FDIM, col0, lane);
      v2f a0 = load_afrag(sT, 0, ST_LD, k0, lane);
      v2f a1 = load_afrag(sT, 16, ST_LD, k0, lane);
      v2f a2 = load_afrag(sT, 32, ST_LD, k0, lane);
      v2f a3 = load_afrag(sT, 48, ST_LD, k0, lane);
      c0 = wmma4(a0, bf, c0);
      c1 = wmma4(a1, bf, c1);
      c2 = wmma4(a2, bf, c2);
      c3 = wmma4(a3, bf, c3);
    }
    __syncthreads();  // everyone done reading T1 before overwrite
    float bb = bf2[col0 + (lane & 15)];
    store_tile<false, true>(sT, ST_LD, 0, col0, lane, c0, bb);
    store_tile<false, true>(sT, ST_LD, 16, col0, lane, c1, bb);
    store_tile<false, true>(sT, ST_LD, 32, col0, lane, c2, bb);
    store_tile<false, true>(sT, ST_LD, 48, col0, lane, c3, bb);
  }
  __syncthreads();

  // ---- aggregation: agg[f] = (1/H) * sum_k mask*W[k,f]*sum_h alpha*y_nb ----
  {
    int f    = tid & 127;
    int half = tid >> 7;
    const float* yb = y + (size_t)(b * HDIM) * NDIM * FDIM;
    float acc = 0.f;
    int kbeg = half * 32;
#pragma unroll 4
    for (int k = kbeg; k < kbeg + 32; ++k) {
      int nb = sNb[k];
      float z = sAtt[0 * KDIM + k] * yb[((size_t)0 * NDIM + nb) * FDIM + f] +
                sAtt[1 * KDIM + k] * yb[((size_t)1 * NDIM + nb) * FDIM + f] +
                sAtt[2 * KDIM + k] * yb[((size_t)2 * NDIM + nb) * FDIM + f] +
                sAtt[3 * KDIM + k] * yb[((size_t)3 * NDIM + nb) * FDIM + f];
      acc += z * sT[k * ST_LD + f] * sMk[k];
    }
    sAgg[half * FDIM + f] = acc;
  }
  __syncthreads();
  if (tid < FDIM) {
    agg[(size_t)bn * FDIM + tid] = (sAgg[tid] + sAgg[FDIM + tid]) * 0.25f;
  }
}

// ---------------------------------------------------------------------------
// Kernel 4: out = ssp(agg(2048x128) * w_out(128x128) + b_out)
// ---------------------------------------------------------------------------
__global__ void __launch_bounds__(128) k_out(const float* __restrict__ agg,
                                             const float* __restrict__ w_out,
                                             const float* __restrict__ b_out,
                                             float* __restrict__ out) {
  int lane = threadIdx.x & 31;
  int wid  = blockIdx.x * 4 + (threadIdx.x >> 5);  // 0..1023
  int row0 = (wid >> 3) << 4;                      // 0..2032
  int col0 = (wid & 7) << 4;                       // 0..112

  v8f c = {};
#pragma unroll 4
  for (int kb = 0; kb < FDIM / 4; ++kb) {
    int k0 = kb * 4;
    v2f af = load_afrag(agg, row0, FDIM, k0, lane);
    v2f bf = load_bfrag(w_out, k0, ODIM, col0, lane);
    c = wmma4(af, bf, c);
  }
  float bb = b_out[col0 + (lane & 15)];
  store_tile<true, true>(out, ODIM, row0, col0, lane, c, bb);
}

// ---------------------------------------------------------------------------
extern "C" void kernel_launch(void* const* d_in, const int* in_sizes, int n_in,
                              void* d_out, int out_size, void* d_ws,
                              size_t ws_size, hipStream_t stream) {
  (void)in_sizes; (void)n_in; (void)out_size; (void)ws_size;
  const float* x      = (const float*)d_in[0];
  // d_in[1] = r_ij (unused by the reference)
  const int*   nbrs   = (const int*)d_in[2];
  const float* mask   = (const float*)d_in[3];
  const float* fij    = (const float*)d_in[4];
  const float* w_in2f = (const float*)d_in[5];
  const float* w_att  = (const float*)d_in[6];
  const float* wf1    = (const float*)d_in[7];
  const float* bf1    = (const float*)d_in[8];
  const float* wf2    = (const float*)d_in[9];
  const float* bf2    = (const float*)d_in[10];
  const float* w_out  = (const float*)d_in[11];
  const float* b_out  = (const float*)d_in[12];
  float* out = (float*)d_out;

  // workspace layout (floats): y | a_c | a_n | agg  (~5.3 MB total)
  float* ws  = (float*)d_ws;
  float* y   = ws;                                          // B*H*N*F
  float* a_c = y + (size_t)BDIM * HDIM * NDIM * FDIM;       // B*H*N
  float* a_n = a_c + (size_t)BDIM * HDIM * NDIM;            // B*H*N
  float* agg = a_n + (size_t)BDIM * HDIM * NDIM;            // B*N*F

  k_proj<<<dim3(1024), dim3(128), 0, stream>>>(x, w_in2f, y);
  k_att<<<dim3(2048), dim3(128), 0, stream>>>(y, w_att, a_c, a_n);
  k_fused<<<dim3(BDIM * NDIM), dim3(256), 0, stream>>>(
      fij, nbrs, mask, wf1, bf1, wf2, bf2, y, a_c, a_n, agg);
  k_out<<<dim3(256), dim3(128), 0, stream>>>(agg, w_out, b_out, out);
}